// RigidNonRigidLoss_56831007261081
// MI455X (gfx1250) — compile-verified
//
#include <hip/hip_runtime.h>
#include <math.h>

typedef __attribute__((ext_vector_type(2))) float v2f;
typedef __attribute__((ext_vector_type(8))) float v8f;

#define B_ 8
#define N_ 4096
#define M_ 4000
#define K_ 5
#define NT_ (N_ / 16)   // 256
#define MT_ (M_ / 16)   // 250

// workspace layout (float offsets)
#define ACC_ALIGN 0
#define ACC_DISP  1
#define ACC_DEF   2
#define ACC_LAP   3
#define ACC_RMSE  4                       // 8 entries (per batch)
#define YR_OFF    16                      // Y_rigid  B*N*3
#define NYR_OFF   (YR_OFF + B_ * N_ * 3)  // |Y_rigid|^2  B*N
#define NX_OFF    (NYR_OFF + B_ * N_)     // |X|^2        B*M
#define NXH_OFF   (NX_OFF + B_ * M_)      // |X_hat|^2    B*N

#define BIGF 3.0e38f

// ---------------- helpers ----------------

__device__ __forceinline__ void ins_val(float d, float t[K_]) {
#pragma unroll
  for (int i = 0; i < K_; ++i) {
    float lo = fminf(d, t[i]);
    float hi = fmaxf(d, t[i]);
    t[i] = lo;
    d = hi;
  }
}

__device__ __forceinline__ void ins_pair(float d, int di, float v[K_], int id[K_]) {
#pragma unroll
  for (int i = 0; i < K_; ++i) {
    bool lt = d < v[i];
    float nv = lt ? d : v[i];
    int   ni = lt ? di : id[i];
    float cd = lt ? v[i] : d;
    int   ci = lt ? id[i] : di;
    v[i] = nv; id[i] = ni; d = cd; di = ci;
  }
}

// ---------------- kernels ----------------

__global__ void init_acc(float* ws) {
  if (threadIdx.x < 16) ws[threadIdx.x] = 0.0f;
}

// per (b,n): Y_rigid, norms, rmse & disp partial sums
__global__ void prep_y(const float* __restrict__ Y, const float* __restrict__ Rp,
                       const float* __restrict__ tp, const float* __restrict__ Rg,
                       const float* __restrict__ tg, const float* __restrict__ Xh,
                       const float* __restrict__ dl, float* __restrict__ ws) {
  int tid = blockIdx.x * blockDim.x + threadIdx.x;  // B*N threads
  int b = tid >> 12;
  const float* y = Y + (size_t)tid * 3;
  float yx = y[0], yy = y[1], yz = y[2];
  const float* rp = Rp + b * 9;
  const float* rg = Rg + b * 9;
  float pr[3], gr[3];
#pragma unroll
  for (int k = 0; k < 3; ++k) {
    pr[k] = rp[3 * k] * yx + rp[3 * k + 1] * yy + rp[3 * k + 2] * yz + tp[b * 3 + k];
    gr[k] = rg[3 * k] * yx + rg[3 * k + 1] * yy + rg[3 * k + 2] * yz + tg[b * 3 + k];
  }
  float* yr = ws + YR_OFF + (size_t)tid * 3;
  yr[0] = pr[0]; yr[1] = pr[1]; yr[2] = pr[2];
  ws[NYR_OFF + tid] = pr[0] * pr[0] + pr[1] * pr[1] + pr[2] * pr[2];
  const float* xh = Xh + (size_t)tid * 3;
  ws[NXH_OFF + tid] = xh[0] * xh[0] + xh[1] * xh[1] + xh[2] * xh[2];
  float d0 = pr[0] - gr[0], d1 = pr[1] - gr[1], d2 = pr[2] - gr[2];
  float rmse = d0 * d0 + d1 * d1 + d2 * d2;
  const float* d = dl + (size_t)tid * 3;
  float disp = d[0] * d[0] + d[1] * d[1] + d[2] * d[2];
#pragma unroll
  for (int off = 16; off >= 1; off >>= 1) {
    rmse += __shfl_down(rmse, off, 32);
    disp += __shfl_down(disp, off, 32);
  }
  if ((threadIdx.x & 31) == 0) {
    atomicAdd(ws + ACC_RMSE + b, rmse);   // whole wave shares one b (N_ % 32 == 0)
    atomicAdd(ws + ACC_DISP, disp);
  }
}

__global__ void prep_x(const float* __restrict__ X, float* __restrict__ ws) {
  int tid = blockIdx.x * blockDim.x + threadIdx.x;  // B*M threads
  const float* x = X + (size_t)tid * 3;
  ws[NX_OFF + tid] = x[0] * x[0] + x[1] * x[1] + x[2] * x[2];
}

// L_align: 5 smallest squared distances X_hat vs X, via v_wmma_f32_16x16x4_f32
__global__ void align_k(const float* __restrict__ X, const float* __restrict__ Xh,
                        float* __restrict__ ws) {
  int gtid = blockIdx.x * blockDim.x + threadIdx.x;
  int wave = gtid >> 5;
  int lane = threadIdx.x & 31;
  int b = wave >> 8;          // / NT_
  int nt = wave & (NT_ - 1);
  int half = lane >> 4;
  int ncol = (nt << 4) + (lane & 15);

  const float* p = Xh + ((size_t)b * N_ + ncol) * 3;
  float px = p[0], py = p[1], pz = p[2];
  v2f bm;
  bm.x = half ? pz : px;
  bm.y = half ? 0.0f : py;
  float nB = ws[NXH_OFF + b * N_ + ncol];
  const float* nx = ws + NX_OFF + b * M_;

  float t[K_];
#pragma unroll
  for (int i = 0; i < K_; ++i) t[i] = BIGF;

  for (int mt = 0; mt < MT_; ++mt) {
    int mrow = (mt << 4) + (lane & 15);
    const float* q = X + ((size_t)b * M_ + mrow) * 3;
    float qx = q[0], qy = q[1], qz = q[2];
    v2f am;
    am.x = half ? qz : qx;
    am.y = half ? 0.0f : qy;
    v8f c = {0.f, 0.f, 0.f, 0.f, 0.f, 0.f, 0.f, 0.f};
    c = __builtin_amdgcn_wmma_f32_16x16x4_f32(false, am, false, bm, (short)0, c,
                                              false, false);
    int mbase = (mt << 4) + (half << 3);
#pragma unroll
    for (int r = 0; r < 8; ++r) {
      float d = nx[mbase + r] + nB - 2.0f * c[r];
      d = fmaxf(d, 0.0f);
      ins_val(d, t);
    }
  }
  // merge the two lane halves (each covered half of the m rows)
  float o[K_];
#pragma unroll
  for (int i = 0; i < K_; ++i) o[i] = __shfl(t[i], (lane & 15) + 16, 32);
#pragma unroll
  for (int i = 0; i < K_; ++i) ins_val(o[i], t);

  float s = (lane < 16) ? (t[0] + t[1] + t[2] + t[3] + t[4]) : 0.0f;
#pragma unroll
  for (int off = 16; off >= 1; off >>= 1) s += __shfl_down(s, off, 32);
  if (lane == 0) atomicAdd(ws + ACC_ALIGN, s);
}

// kNN on Y_rigid (self-excluded) + deform & laplacian losses
__global__ void knn_k(const float* __restrict__ Xh, const float* __restrict__ dl,
                      float* __restrict__ ws) {
  int gtid = blockIdx.x * blockDim.x + threadIdx.x;
  int wave = gtid >> 5;
  int lane = threadIdx.x & 31;
  int b = wave >> 8;
  int nt = wave & (NT_ - 1);
  int half = lane >> 4;
  int ncol = (nt << 4) + (lane & 15);

  const float* yrb = ws + YR_OFF + (size_t)b * N_ * 3;
  const float* ny = ws + NYR_OFF + b * N_;

  const float* p = yrb + (size_t)ncol * 3;
  float px = p[0], py = p[1], pz = p[2];
  v2f bm;
  bm.x = half ? pz : px;
  bm.y = half ? 0.0f : py;
  float nB = ny[ncol];

  float v[K_];
  int id[K_];
#pragma unroll
  for (int i = 0; i < K_; ++i) { v[i] = BIGF; id[i] = 0; }

  for (int mt = 0; mt < NT_; ++mt) {
    int mrow = (mt << 4) + (lane & 15);
    const float* q = yrb + (size_t)mrow * 3;
    float qx = q[0], qy = q[1], qz = q[2];
    v2f am;
    am.x = half ? qz : qx;
    am.y = half ? 0.0f : qy;
    v8f c = {0.f, 0.f, 0.f, 0.f, 0.f, 0.f, 0.f, 0.f};
    c = __builtin_amdgcn_wmma_f32_16x16x4_f32(false, am, false, bm, (short)0, c,
                                              false, false);
    int mbase = (mt << 4) + (half << 3);
#pragma unroll
    for (int r = 0; r < 8; ++r) {
      int mid = mbase + r;
      float d = ny[mid] + nB - 2.0f * c[r];
      d = fmaxf(d, 0.0f);
      d = (mid == ncol) ? BIGF : d;  // exclude self
      ins_pair(d, mid, v, id);
    }
  }
  // merge halves
  float ov[K_];
  int oi[K_];
#pragma unroll
  for (int i = 0; i < K_; ++i) {
    ov[i] = __shfl(v[i], (lane & 15) + 16, 32);
    oi[i] = __shfl(id[i], (lane & 15) + 16, 32);
  }
#pragma unroll
  for (int i = 0; i < K_; ++i) ins_pair(ov[i], oi[i], v, id);

  float def = 0.0f, ls = 0.0f;
  if (lane < 16) {
    const float* xhb = Xh + (size_t)b * N_ * 3;
    const float* dlb = dl + (size_t)b * N_ * 3;
    int g = ncol;
    float ycx = yrb[3 * g], ycy = yrb[3 * g + 1], ycz = yrb[3 * g + 2];
    float xcx = xhb[3 * g], xcy = xhb[3 * g + 1], xcz = xhb[3 * g + 2];
    float dcx = dlb[3 * g], dcy = dlb[3 * g + 1], dcz = dlb[3 * g + 2];
    float sx = 0.f, sy = 0.f, sz = 0.f;
#pragma unroll
    for (int k = 0; k < K_; ++k) {
      int j = id[k];
      float ex = (xhb[3 * j] - xcx) - (yrb[3 * j] - ycx);
      float ey = (xhb[3 * j + 1] - xcy) - (yrb[3 * j + 1] - ycy);
      float ez = (xhb[3 * j + 2] - xcz) - (yrb[3 * j + 2] - ycz);
      def += ex * ex + ey * ey + ez * ez;
      sx += dlb[3 * j]; sy += dlb[3 * j + 1]; sz += dlb[3 * j + 2];
    }
    float lx = dcx - sx * 0.2f, ly = dcy - sy * 0.2f, lz = dcz - sz * 0.2f;
    ls = lx * lx + ly * ly + lz * lz;
  }
#pragma unroll
  for (int off = 16; off >= 1; off >>= 1) {
    def += __shfl_down(def, off, 32);
    ls += __shfl_down(ls, off, 32);
  }
  if (lane == 0) {
    atomicAdd(ws + ACC_DEF, def);
    atomicAdd(ws + ACC_LAP, ls);
  }
}

__global__ void final_k(const float* __restrict__ Rp, const float* __restrict__ tp,
                        const float* __restrict__ Rg, const float* __restrict__ tg,
                        const float* __restrict__ ws, float* __restrict__ out) {
  if (threadIdx.x != 0 || blockIdx.x != 0) return;
  float Lrot = 0.f, Ltr = 0.f, Lrmse = 0.f;
  for (int b = 0; b < B_; ++b) {
    const float* rp = Rp + b * 9;
    const float* rg = Rg + b * 9;
    float tr = 0.f;
    for (int i = 0; i < 9; ++i) tr += rp[i] * rg[i];  // trace(Rp^T Rg)
    float ca = (tr - 1.0f) * 0.5f;
    ca = fminf(fmaxf(ca, -1.0f + 1e-7f), 1.0f - 1e-7f);
    Lrot += acosf(ca);
    float d0 = tp[b * 3] - tg[b * 3];
    float d1 = tp[b * 3 + 1] - tg[b * 3 + 1];
    float d2 = tp[b * 3 + 2] - tg[b * 3 + 2];
    Ltr += sqrtf(d0 * d0 + d1 * d1 + d2 * d2);
    Lrmse += sqrtf(ws[ACC_RMSE + b] / (float)N_);
  }
  Lrot *= (1.0f / B_);
  Ltr *= (1.0f / B_);
  Lrmse *= (1.0f / B_);
  float La   = ws[ACC_ALIGN] / ((float)B_ * N_ * K_);
  float Ld   = ws[ACC_DISP]  / ((float)B_ * N_);
  float Ldef = ws[ACC_DEF]   / ((float)B_ * N_ * K_);
  float Llap = ws[ACC_LAP]   / ((float)B_ * N_);
  float rigid = Lrot + Ltr + Lrmse;
  float nonrigid = La + 0.01f * Ld + 0.1f * Ldef + 0.1f * Llap;
  out[0] = rigid + nonrigid;
  out[1] = rigid;
  out[2] = nonrigid;
  out[3] = Lrot;
  out[4] = Ltr;
  out[5] = Lrmse;
  out[6] = La;
  out[7] = Ld;
  out[8] = Ldef;
  out[9] = Llap;
}

extern "C" void kernel_launch(void* const* d_in, const int* in_sizes, int n_in,
                              void* d_out, int out_size, void* d_ws, size_t ws_size,
                              hipStream_t stream) {
  (void)in_sizes; (void)n_in; (void)out_size; (void)ws_size;
  const float* Y  = (const float*)d_in[0];
  const float* X  = (const float*)d_in[1];
  const float* Rp = (const float*)d_in[2];
  const float* tp = (const float*)d_in[3];
  const float* Rg = (const float*)d_in[4];
  const float* tg = (const float*)d_in[5];
  const float* Xh = (const float*)d_in[6];
  const float* dl = (const float*)d_in[7];
  float* ws = (float*)d_ws;
  float* out = (float*)d_out;

  hipLaunchKernelGGL(init_acc, dim3(1), dim3(32), 0, stream, ws);
  hipLaunchKernelGGL(prep_y, dim3(B_ * N_ / 256), dim3(256), 0, stream,
                     Y, Rp, tp, Rg, tg, Xh, dl, ws);
  hipLaunchKernelGGL(prep_x, dim3(B_ * M_ / 256), dim3(256), 0, stream, X, ws);
  hipLaunchKernelGGL(align_k, dim3(B_ * NT_ / 8), dim3(256), 0, stream, X, Xh, ws);
  hipLaunchKernelGGL(knn_k, dim3(B_ * NT_ / 8), dim3(256), 0, stream, Xh, dl, ws);
  hipLaunchKernelGGL(final_k, dim3(1), dim3(32), 0, stream, Rp, tp, Rg, tg, ws, out);
}